// iterative_module_90271622627339
// MI455X (gfx1250) — compile-verified
//
#include <hip/hip_runtime.h>

// ---------------- problem constants (from reference) ----------------
#define DD 256
#define HH 256
#define WW 256
#define NPROJ 8
#define NV 192
#define NU 192
#define N_STEPS 448
#define NRAYS (NPROJ * NV * NU)   // 294912 == 1152 * 256 (exact multiple of block)
#define HW (HH * WW)              // 65536

typedef __attribute__((ext_vector_type(2))) float v2f;
typedef __attribute__((ext_vector_type(8))) float v8f;

// -------- prep: invert 3x3 of each 3x4 projection matrix, compute src --------
// ws layout: per proj, 16 floats: [0..8]=Minv row-major, [9..11]=src, rest unused
__global__ void cone_prep_kernel(const float* __restrict__ pm, float* __restrict__ ws) {
    int p = threadIdx.x;
    if (p >= NPROJ) return;
    const float* P = pm + p * 12;
    float m00 = P[0], m01 = P[1], m02 = P[2],  b0 = P[3];
    float m10 = P[4], m11 = P[5], m12 = P[6],  b1 = P[7];
    float m20 = P[8], m21 = P[9], m22 = P[10], b2 = P[11];
    float c00 = m11 * m22 - m12 * m21;
    float c01 = -(m10 * m22 - m12 * m20);
    float c02 = m10 * m21 - m11 * m20;
    float det = m00 * c00 + m01 * c01 + m02 * c02;
    float id  = 1.0f / det;
    float i00 = c00 * id;
    float i01 = (m02 * m21 - m01 * m22) * id;
    float i02 = (m01 * m12 - m02 * m11) * id;
    float i10 = c01 * id;
    float i11 = (m00 * m22 - m02 * m20) * id;
    float i12 = (m02 * m10 - m00 * m12) * id;
    float i20 = c02 * id;
    float i21 = (m01 * m20 - m00 * m21) * id;
    float i22 = (m00 * m11 - m01 * m10) * id;
    float* w = ws + p * 16;
    w[0] = i00; w[1] = i01; w[2] = i02;
    w[3] = i10; w[4] = i11; w[5] = i12;
    w[6] = i20; w[7] = i21; w[8] = i22;
    w[9]  = -(i00 * b0 + i01 * b1 + i02 * b2);
    w[10] = -(i10 * b0 + i11 * b1 + i12 * b2);
    w[11] = -(i20 * b0 + i21 * b1 + i22 * b2);
}

// -------- main: one ray per lane; directions via fp32 WMMA; march 448 steps --------
__global__ __launch_bounds__(256) void cone_fp_kernel(const float* __restrict__ vol,
                                                      const float* __restrict__ ws,
                                                      float* __restrict__ sino) {
    const int ray  = blockIdx.x * 256 + threadIdx.x;          // grid is exact, no tail
    const int lane = threadIdx.x & 31;
    const int p    = __builtin_amdgcn_readfirstlane(ray / (NU * NV)); // uniform per wave
    const int rem  = ray % (NU * NV);
    const int v    = rem / NU;                                 // uniform per wave (32|192)
    const int u0   = __builtin_amdgcn_readfirstlane((rem % NU) - lane); // wave's first u
    const float vf = (float)v;

    const float* wp = ws + p * 16;

    // ---- ray direction d = Minv * [u, v, 1]^T via V_WMMA_F32_16X16X4_F32 ----
    // A (16x4): rows 0..2 = Minv rows, rows 8..10 = Minv rows (duplicated so
    // lanes 16..31 of D, which expose rows 8..15, also carry the result).
    // B (4x16): columns = 16 rays: [u, v, 1, 0]^T.  Two WMMAs cover 32 rays.
    const int  r   = lane & 15;
    const bool hi  = (lane >= 16);
    const bool rA  = (r < 3);
    const bool rB  = (r >= 8) && (r < 11);
    const int  rr  = rA ? r : (rB ? (r - 8) : 0);
    const float sel = (rA || rB) ? 1.0f : 0.0f;

    v2f A;                                   // lane<16: A[r][0],A[r][1]; lane>=16: A[r][2],A[r][3]
    A.x = wp[rr * 3 + (hi ? 2 : 0)] * sel;
    A.y = hi ? 0.0f : (wp[rr * 3 + 1] * sel);

    v2f B1, B2;                              // lane<16: K rows 0,1; lane>=16: K rows 2,3
    B1.x = hi ? 1.0f : (float)(u0 + r);
    B1.y = hi ? 0.0f : vf;
    B2.x = hi ? 1.0f : (float)(u0 + 16 + r);
    B2.y = hi ? 0.0f : vf;

    v8f cz = {};
    v8f D1 = __builtin_amdgcn_wmma_f32_16x16x4_f32(false, A, false, B1, (short)0, cz, false, false);
    v8f D2 = __builtin_amdgcn_wmma_f32_16x16x4_f32(false, A, false, B2, (short)0, cz, false, false);

    // lane L<16 reads rows 0..2 (rays u0+L); lane L>=16 reads rows 8..10 of D2 (rays u0+L)
    float dx = hi ? D2[0] : D1[0];
    float dy = hi ? D2[1] : D1[1];
    float dz = hi ? D2[2] : D1[2];

    float nrm = sqrtf(dx * dx + dy * dy + dz * dz);
    dx /= nrm; dy /= nrm; dz /= nrm;

    const float sx = wp[9], sy = wp[10], sz = wp[11];

    // slab intersection (matches reference: dsafe = |d|<1e-9 ? 1e-9 : d)
    const float BMIN = -127.5f, BMAX = 127.5f;
    float ddx = (fabsf(dx) < 1e-9f) ? 1e-9f : dx;
    float ddy = (fabsf(dy) < 1e-9f) ? 1e-9f : dy;
    float ddz = (fabsf(dz) < 1e-9f) ? 1e-9f : dz;
    float t1x = (BMIN - sx) / ddx, t2x = (BMAX - sx) / ddx;
    float t1y = (BMIN - sy) / ddy, t2y = (BMAX - sy) / ddy;
    float t1z = (BMIN - sz) / ddz, t2z = (BMAX - sz) / ddz;
    float tnear = fmaxf(fmaxf(fminf(t1x, t2x), fminf(t1y, t2y)), fminf(t1z, t2z));
    tnear = fmaxf(tnear, 0.0f);
    // exit of box enlarged by 1 voxel: beyond this every trilinear sample is exactly 0,
    // so breaking does not change the sum vs the reference's fixed 448 steps.
    float e1x = (BMIN - 1.0f - sx) / ddx, e2x = (BMAX + 1.0f - sx) / ddx;
    float e1y = (BMIN - 1.0f - sy) / ddy, e2y = (BMAX + 1.0f - sy) / ddy;
    float e1z = (BMIN - 1.0f - sz) / ddz, e2z = (BMAX + 1.0f - sz) / ddz;
    float texit = fminf(fminf(fmaxf(e1x, e2x), fmaxf(e1y, e2y)), fmaxf(e1z, e2z));

    // continuous voxel coords: c = (s - ORIGIN) + t*d  (SPACING = 1, ORIGIN = -127.5)
    const float sox = sx + 127.5f, soy = sy + 127.5f, soz = sz + 127.5f;

    // prefetch offset ~8 steps ahead along the ray
    const int pfx = (int)floorf(dx * 8.0f + 0.5f);
    const int pfy = (int)floorf(dy * 8.0f + 0.5f);
    const int pfz = (int)floorf(dz * 8.0f + 0.5f);
    const long pfo = (long)pfz * HW + (long)pfy * WW + pfx;

    float acc = 0.0f;
    for (int i = 0; i < N_STEPS; ++i) {
        float t = tnear + (float)i;
        if (t > texit) break;
        float cx = fmaf(t, dx, sox);
        float cy = fmaf(t, dy, soy);
        float czc = fmaf(t, dz, soz);
        float fx0 = floorf(cx), fy0 = floorf(cy), fz0 = floorf(czc);
        int x0 = (int)fx0, y0 = (int)fy0, z0 = (int)fz0;
        float fx = cx - fx0, fy = cy - fy0, fz = czc - fz0;

        float s000, s001, s010, s011, s100, s101, s110, s111;
        if (((unsigned)x0 < (unsigned)(WW - 1)) &
            ((unsigned)y0 < (unsigned)(HH - 1)) &
            ((unsigned)z0 < (unsigned)(DD - 1))) {
            // fast path: all 8 corners in-bounds
            const float* bp = vol + ((size_t)z0 * HW + (size_t)y0 * WW + x0);
            __builtin_prefetch(bp + pfo, 0, 1);
            s000 = bp[0];           s001 = bp[1];
            s010 = bp[WW];          s011 = bp[WW + 1];
            s100 = bp[HW];          s101 = bp[HW + 1];
            s110 = bp[HW + WW];     s111 = bp[HW + WW + 1];
        } else {
            // boundary path: clamp index, mask value (matches reference samp())
            auto samp = [&](int ix, int iy, int iz) -> float {
                bool ok = ((unsigned)ix < (unsigned)WW) &
                          ((unsigned)iy < (unsigned)HH) &
                          ((unsigned)iz < (unsigned)DD);
                int cxi = ix < 0 ? 0 : (ix > WW - 1 ? WW - 1 : ix);
                int cyi = iy < 0 ? 0 : (iy > HH - 1 ? HH - 1 : iy);
                int czi = iz < 0 ? 0 : (iz > DD - 1 ? DD - 1 : iz);
                float val = vol[(size_t)czi * HW + (size_t)cyi * WW + cxi];
                return ok ? val : 0.0f;
            };
            int x1 = x0 + 1, y1 = y0 + 1, z1 = z0 + 1;
            s000 = samp(x0, y0, z0); s001 = samp(x1, y0, z0);
            s010 = samp(x0, y1, z0); s011 = samp(x1, y1, z0);
            s100 = samp(x0, y0, z1); s101 = samp(x1, y0, z1);
            s110 = samp(x0, y1, z1); s111 = samp(x1, y1, z1);
        }
        float c00 = fmaf(fx, s001 - s000, s000);
        float c10 = fmaf(fx, s011 - s010, s010);
        float c01 = fmaf(fx, s101 - s100, s100);
        float c11 = fmaf(fx, s111 - s110, s110);
        float c0  = fmaf(fy, c10 - c00, c00);
        float c1  = fmaf(fy, c11 - c01, c01);
        acc += fmaf(fz, c1 - c0, c0);
    }
    sino[ray] = acc;   // * STEP (=1.0)
}

extern "C" void kernel_launch(void* const* d_in, const int* in_sizes, int n_in,
                              void* d_out, int out_size, void* d_ws, size_t ws_size,
                              hipStream_t stream) {
    const float* reco = (const float*)d_in[0];   // [1,256,256,256] fp32
    const float* pm   = (const float*)d_in[1];   // [8,3,4] fp32
    float* out = (float*)d_out;                  // sino (294912) ++ reco (16777216)
    float* ws  = (float*)d_ws;                   // needs 8*16 floats = 512 B

    // pass-through output: reco
    hipMemcpyAsync(out + NRAYS, reco, (size_t)DD * HH * WW * sizeof(float),
                   hipMemcpyDeviceToDevice, stream);

    cone_prep_kernel<<<1, NPROJ, 0, stream>>>(pm, ws);
    cone_fp_kernel<<<NRAYS / 256, 256, 0, stream>>>(reco, ws, out);
}